// HungarianLoss_38336878084438
// MI455X (gfx1250) — compile-verified
//
#include <hip/hip_runtime.h>
#include <math.h>

// ---------------------------------------------------------------------------
// HungarianLoss for MI455X (gfx1250):
//   phase 1: row norms of pred/target            (wave-per-row, shfl reduce)
//   phase 2: dist = sqrt(|p|^2+|t|^2-2 p.t)      (fp32 WMMA 16x16x4 per wave)
//   phase 3: Jonker-Volgenant LSA per batch      (single-wave WG, 256KB cost
//            matrix fully resident in CDNA5's 320KB LDS)
//   phase 4: deterministic final mean
// ---------------------------------------------------------------------------

typedef __attribute__((ext_vector_type(2))) float v2f;
typedef __attribute__((ext_vector_type(8))) float v8f;

#define BB 16
#define LL 256
#define DD 128

// ---------------- phase 1: squared row norms -------------------------------
__global__ __launch_bounds__(256) void norms_kernel(const float* __restrict__ pred,
                                                    const float* __restrict__ target,
                                                    float* __restrict__ pn2,
                                                    float* __restrict__ tn2) {
  int wave = (blockIdx.x * blockDim.x + threadIdx.x) >> 5;  // 2*B*L waves
  int lane = threadIdx.x & 31;
  const float* src;
  float* dst;
  int row = wave;
  if (row < BB * LL) {
    src = pred + (size_t)row * DD;
    dst = pn2 + row;
  } else {
    row -= BB * LL;
    src = target + (size_t)row * DD;
    dst = tn2 + row;
  }
  float4 x = ((const float4*)src)[lane];  // 32 lanes * 4 = 128 elements
  float s = x.x * x.x + x.y * x.y + x.z * x.z + x.w * x.w;
  #pragma unroll
  for (int o = 16; o > 0; o >>= 1) s += __shfl_xor(s, o, 32);
  if (lane == 0) *dst = s;
}

// ---------------- phase 2: pairwise distances via fp32 WMMA ----------------
// One wave computes one 16x16 tile of dist[b].  A = pred rows (16xK),
// B = target rows read as KxN (element (k,n) = target[n*D+k]).
// fp32 WMMA layouts (ISA 7.12.2): A 16x4: lane m(0..15) holds K pair at
// koff = 2*(lane>=16); B 4x16 mirrors it; C/D 8 VGPRs, M = r + 8*(lane>=16),
// N = lane&15.
__global__ __launch_bounds__(256) void dist_wmma_kernel(const float* __restrict__ pred,
                                                        const float* __restrict__ target,
                                                        const float* __restrict__ pn2,
                                                        const float* __restrict__ tn2,
                                                        float* __restrict__ dist) {
  int wave = (blockIdx.x * blockDim.x + threadIdx.x) >> 5;  // B*16*16 waves
  int lane = threadIdx.x & 31;
  int b = wave >> 8;
  int tile = wave & 255;
  int row0 = (tile >> 4) << 4;
  int col0 = (tile & 15) << 4;

  const float* Ap = pred + ((size_t)b * LL + row0) * DD;
  const float* Bp = target + ((size_t)b * LL + col0) * DD;
  int m = lane & 15;         // A row index == B column index
  int koff = (lane >> 4) * 2;

  v8f c = {};
  #pragma unroll 4
  for (int k = 0; k < DD; k += 4) {
    v2f a = *(const v2f*)(Ap + (size_t)m * DD + k + koff);
    v2f bmat = *(const v2f*)(Bp + (size_t)m * DD + k + koff);
    c = __builtin_amdgcn_wmma_f32_16x16x4_f32(false, a, false, bmat,
                                              (short)0, c, false, false);
  }

  const float* pr = pn2 + b * LL + row0;
  float tn = tn2[b * LL + col0 + m];
  int half = lane >> 4;
  float* dbase = dist + (size_t)b * LL * LL;
  #pragma unroll
  for (int r = 0; r < 8; ++r) {
    int mm = r + 8 * half;
    float val = pr[mm] + tn - 2.0f * c[r];
    val = (val > 0.0f) ? sqrtf(val) : 0.0f;
    dbase[(size_t)(row0 + mm) * LL + col0 + m] = val;
  }
}

// ---------------- phase 3: Jonker-Volgenant assignment ---------------------
// One 32-thread (single-wave) workgroup per batch: __syncthreads compiles to
// a bare LDS-wait (no barrier resource for single-wave WGs), and the whole
// 256KB cost matrix lives in CDNA5's 320KB LDS.
__global__ __launch_bounds__(32) void hungarian_kernel(const float* __restrict__ dist,
                                                       float* __restrict__ partial) {
  __shared__ __align__(16) float costS[LL * LL];   // 256 KB
  __shared__ float u[LL + 2], v[LL + 2], minv[LL + 2];
  __shared__ int way[LL + 2], p[LL + 2], usedS[LL + 2];

  const int b = blockIdx.x;
  const int t = threadIdx.x;
  const float* cost = dist + (size_t)b * LL * LL;

  for (int i = t; i < (LL * LL) / 4; i += 32)
    ((float4*)costS)[i] = ((const float4*)cost)[i];
  for (int j = t; j <= LL; j += 32) { u[j] = 0.0f; v[j] = 0.0f; p[j] = 0; way[j] = 0; }
  __syncthreads();

  for (int i = 1; i <= LL; ++i) {
    for (int j = t; j <= LL; j += 32) { minv[j] = INFINITY; usedS[j] = 0; }
    if (t == 0) p[0] = i;
    __syncthreads();

    int j0 = 0;
    while (true) {
      if (t == 0) usedS[j0] = 1;
      __syncthreads();
      int i0 = p[j0];
      float ui0 = u[i0];
      const float* crow = costS + (size_t)(i0 - 1) * LL;

      // update minv over free columns; build (orderable-key | index) argmin
      unsigned long long best = ~0ULL;
      for (int j = t + 1; j <= LL; j += 32) {
        if (!usedS[j]) {
          float cur = crow[j - 1] - ui0 - v[j];
          float mv = minv[j];
          if (cur < mv) { mv = cur; minv[j] = cur; way[j] = j0; }
          unsigned kb = __float_as_uint(mv);
          kb = (kb & 0x80000000u) ? ~kb : (kb | 0x80000000u);  // monotonic map
          unsigned long long pk = ((unsigned long long)kb << 32) | (unsigned)j;
          if (pk < best) best = pk;  // low bits = index -> first-min tie-break
        }
      }
      #pragma unroll
      for (int o = 16; o > 0; o >>= 1) {
        unsigned long long other = __shfl_xor(best, o, 32);
        if (other < best) best = other;
      }
      int j1 = (int)(best & 0xFFFFFFFFu);
      unsigned kb = (unsigned)(best >> 32);
      unsigned fb = (kb & 0x80000000u) ? (kb ^ 0x80000000u) : ~kb;
      float delta = __uint_as_float(fb);
      __syncthreads();

      // potential updates: matched rows are distinct -> conflict-free
      for (int jj = t; jj <= LL; jj += 32) {
        if (usedS[jj]) { u[p[jj]] += delta; v[jj] -= delta; }
        else           { minv[jj] -= delta; }
      }
      __syncthreads();
      j0 = j1;
      if (p[j0] == 0) break;
    }

    if (t == 0) {  // augment along the alternating path
      int j = j0;
      while (j != 0) { int jn = way[j]; p[j] = p[jn]; j = jn; }
    }
    __syncthreads();
  }

  // per-batch matched-cost sum: row p[j]-1 is matched to column j-1
  float s = 0.0f;
  for (int j = t + 1; j <= LL; j += 32) s += costS[(size_t)(p[j] - 1) * LL + (j - 1)];
  #pragma unroll
  for (int o = 16; o > 0; o >>= 1) s += __shfl_xor(s, o, 32);
  if (t == 0) partial[b] = s;
}

// ---------------- phase 4: deterministic final mean ------------------------
__global__ void final_kernel(const float* __restrict__ partial, float* __restrict__ out) {
  if (threadIdx.x == 0 && blockIdx.x == 0) {
    float s = 0.0f;
    for (int i = 0; i < BB; ++i) s += partial[i];
    out[0] = s / (float)(BB * LL);
  }
}

// ---------------------------------------------------------------------------
extern "C" void kernel_launch(void* const* d_in, const int* in_sizes, int n_in,
                              void* d_out, int out_size, void* d_ws, size_t ws_size,
                              hipStream_t stream) {
  const float* pred = (const float*)d_in[0];
  const float* target = (const float*)d_in[1];

  float* ws = (float*)d_ws;
  float* pn2 = ws;                        // B*L
  float* tn2 = pn2 + BB * LL;             // B*L
  float* dist = tn2 + BB * LL;            // B*L*L (4 MB)
  float* partial = dist + (size_t)BB * LL * LL;  // B

  // phase 1: 2*B*L waves, 8 waves per 256-thread block
  norms_kernel<<<(2 * BB * LL) / 8, 256, 0, stream>>>(pred, target, pn2, tn2);
  // phase 2: B*16*16 = 4096 tile-waves, 8 per block
  dist_wmma_kernel<<<(BB * 256) / 8, 256, 0, stream>>>(pred, target, pn2, tn2, dist);
  // phase 3: one single-wave workgroup per batch
  hungarian_kernel<<<BB, 32, 0, stream>>>(dist, partial);
  // phase 4
  final_kernel<<<1, 1, 0, stream>>>(partial, (float*)d_out);
}